// RoutingCapsules_40819369181602
// MI455X (gfx1250) — compile-verified
//
#include <hip/hip_runtime.h>
#include <hip/hip_bf16.h>

// Problem constants (from reference setup_inputs)
#define CAP_B 256
#define CAP_K 10
#define CAP_I 1152
#define CAP_O 16
#define CAP_C 8

typedef __attribute__((ext_vector_type(2))) float v2f;
typedef __attribute__((ext_vector_type(8))) float v8f;

// ---------------------------------------------------------------------------
// Kernel 1: u_hat[b,k,i,o] = sum_c w[k,i,o,c] * x[b,i,c]
// For each (k,i): U[16x16] = Xtile[16 x 8] * W^T[8 x 16] via two
// V_WMMA_F32_16X16X4_F32 ops (exact fp32, contraction split K=0..3, 4..7).
//
// Per ISA 7.12.2 (wave32):
//   A (16x4 f32):  lanes 0-15 -> M=lane, VGPR j -> K = (lane/16)*2 + j
//   B (4x16 f32):  lanes -> N = lane%16, VGPR j -> K = (lane/16)*2 + j
//   D (16x16 f32): VGPR r -> M = r + 8*(lane/16), N = lane%16
// ---------------------------------------------------------------------------
__global__ __launch_bounds__(256)
void capsule_uhat_wmma(const float* __restrict__ x,
                       const float* __restrict__ w,
                       float* __restrict__ uhat,   // [chunkB, K, I, O]
                       int b0, int chunkB) {
    const int ki   = blockIdx.x;           // 0 .. K*I-1
    const int k    = ki / CAP_I;
    const int i    = ki % CAP_I;
    const int wave = threadIdx.x >> 5;
    const int lane = threadIdx.x & 31;

    const int o    = lane & 15;            // N column (output capsule dim)
    const int half = lane >> 4;            // 0: K rows {0,1}/{4,5}; 1: {2,3}/{6,7}
    const int cbase = half << 1;           // contraction base for this lane

    // B fragment (shared across all batch tiles for this (k,i))
    const float* wb = w + (((size_t)k * CAP_I + i) * CAP_O + o) * CAP_C + cbase;
    const v2f bfrag_lo = *(const v2f*)(wb);       // c = cbase, cbase+1
    const v2f bfrag_hi = *(const v2f*)(wb + 4);   // c = cbase+4, cbase+5

    const int ntile = chunkB >> 4;         // 16-batch tiles in this chunk
    for (int t = wave; t < ntile; t += 8) {
        const int brow = t * 16 + (lane & 15);              // local batch row (A: M=lane%16)
        const float* xb = x + ((size_t)(b0 + brow) * CAP_I + i) * CAP_C + cbase;
        const v2f afrag_lo = *(const v2f*)(xb);
        const v2f afrag_hi = *(const v2f*)(xb + 4);

        v8f acc = {};
        acc = __builtin_amdgcn_wmma_f32_16x16x4_f32(
                  false, afrag_lo, false, bfrag_lo, (short)0, acc, false, false);
        acc = __builtin_amdgcn_wmma_f32_16x16x4_f32(
                  false, afrag_hi, false, bfrag_hi, (short)0, acc, false, false);

        // Scatter D: row r -> batch t*16 + r + 8*half, col -> o
        #pragma unroll
        for (int r = 0; r < 8; ++r) {
            const int bm = t * 16 + r + half * 8;           // local batch index
            uhat[(((size_t)bm * CAP_K + k) * CAP_I + i) * CAP_O + o] = acc[r];
        }
    }
}

// ---------------------------------------------------------------------------
// Kernel 2: dynamic routing for one batch element per workgroup.
// LDS: logits[K][I] + c[K][I] + s[K][O] + v[K][O]  (~93 KB, fits 320 KB WGP LDS)
// ---------------------------------------------------------------------------
__global__ __launch_bounds__(256)
void capsule_route(const float* __restrict__ uhat,  // [chunkB, K, I, O]
                   float* __restrict__ out,         // [B, K, O]
                   int b0) {
    extern __shared__ float smem[];
    float* logits = smem;                       // K*I
    float* cmat   = logits + CAP_K * CAP_I;     // K*I
    float* s_lds  = cmat + CAP_K * CAP_I;       // K*O
    float* v_lds  = s_lds + CAP_K * CAP_O;      // K*O

    const int tid = threadIdx.x;
    const int bl  = blockIdx.x;                 // local batch index in chunk
    const float* ub = uhat + (size_t)bl * CAP_K * CAP_I * CAP_O;

    const int k = tid >> 4;                     // valid when tid < 160
    const int o = tid & 15;

    for (int idx = tid; idx < CAP_K * CAP_I; idx += 256) logits[idx] = 0.0f;
    __syncthreads();

    for (int it = 0; it < 3; ++it) {
        // c = softmax over K (per i)
        for (int i = tid; i < CAP_I; i += 256) {
            float m = logits[i];
            #pragma unroll
            for (int kk = 1; kk < CAP_K; ++kk) m = fmaxf(m, logits[kk * CAP_I + i]);
            float e[CAP_K];
            float sum = 0.0f;
            #pragma unroll
            for (int kk = 0; kk < CAP_K; ++kk) {
                e[kk] = __expf(logits[kk * CAP_I + i] - m);
                sum += e[kk];
            }
            const float inv = 1.0f / sum;
            #pragma unroll
            for (int kk = 0; kk < CAP_K; ++kk) cmat[kk * CAP_I + i] = e[kk] * inv;
        }
        __syncthreads();

        // s[k,o] = sum_i c[k,i] * u_hat[k,i,o]   (thread-private full reduction)
        if (tid < CAP_K * CAP_O) {
            const float* up = ub + (size_t)k * CAP_I * CAP_O + o;
            const float* cp = cmat + k * CAP_I;
            float ssum = 0.0f;
            for (int i = 0; i < CAP_I; ++i) {
                if ((i & 31) == 0)
                    __builtin_prefetch(up + (size_t)(i + 64) * CAP_O, 0, 1);
                ssum += cp[i] * up[(size_t)i * CAP_O];
            }
            s_lds[k * CAP_O + o] = ssum;
        }
        __syncthreads();

        // v = squash(s)
        if (tid < CAP_K * CAP_O) {
            float n2 = 0.0f;
            #pragma unroll
            for (int oo = 0; oo < CAP_O; ++oo) {
                const float t = s_lds[k * CAP_O + oo];
                n2 += t * t;
            }
            const float n = sqrtf(n2);
            const float scale = n2 / ((1.0f + n2) * (n + 1e-9f));
            v_lds[k * CAP_O + o] = scale * s_lds[k * CAP_O + o];
        }
        __syncthreads();

        // logits[k,i] += dot_o(u_hat[k,i,:], v[k,:])   (skip on last iteration)
        if (it < 2) {
            for (int p = tid; p < CAP_K * CAP_I; p += 256) {
                const float* up = ub + (size_t)p * CAP_O;   // p = k*I + i
                const int kk = p / CAP_I;
                float dot = 0.0f;
                #pragma unroll
                for (int oo = 0; oo < CAP_O; ++oo) dot += up[oo] * v_lds[kk * CAP_O + oo];
                logits[p] += dot;
            }
            __syncthreads();
        }
    }

    if (tid < CAP_K * CAP_O)
        out[((size_t)(b0 + bl) * CAP_K + k) * CAP_O + o] = v_lds[k * CAP_O + o];
}

// ---------------------------------------------------------------------------
// Host launcher: chunk the batch so u_hat fits in d_ws; stream-order the
// (WMMA, routing) pairs. Deterministic, graph-capture safe.
// ---------------------------------------------------------------------------
extern "C" void kernel_launch(void* const* d_in, const int* in_sizes, int n_in,
                              void* d_out, int out_size, void* d_ws, size_t ws_size,
                              hipStream_t stream) {
    const float* x = (const float*)d_in[0];   // [B, I, C]
    const float* w = (const float*)d_in[1];   // [K, I, O, C]
    float* out = (float*)d_out;               // [B, K, O]
    float* uhat = (float*)d_ws;

    const size_t bytes_per_b = (size_t)CAP_K * CAP_I * CAP_O * sizeof(float); // 737280
    int chunk = (int)(ws_size / bytes_per_b);
    if (chunk > CAP_B) chunk = CAP_B;
    chunk &= ~15;                 // multiple of 16 (WMMA M tile)
    if (chunk < 16) chunk = 16;   // minimum workable chunk

    const size_t smem = (size_t)(2 * CAP_K * CAP_I + 2 * CAP_K * CAP_O) * sizeof(float);

    for (int b0 = 0; b0 < CAP_B; b0 += chunk) {
        int cb = CAP_B - b0;
        if (cb > chunk) cb = chunk;
        capsule_uhat_wmma<<<dim3(CAP_K * CAP_I), dim3(256), 0, stream>>>(
            x, w, uhat, b0, cb);
        capsule_route<<<dim3(cb), dim3(256), smem, stream>>>(uhat, out, b0);
    }
}